// GCN_39256001086159
// MI455X (gfx1250) — compile-verified
//
#include <hip/hip_runtime.h>
#include <hip/hip_bf16.h>

typedef float v2f __attribute__((ext_vector_type(2)));
typedef float v8f __attribute__((ext_vector_type(8)));

#define FEAT 64
#define FILL_WEIGHT 1.0f   // RENORM=True, IMPROVED=False

// ---------------------------------------------------------------------------
// 1) deg[i] = fill_weight  (self-loop contribution)
__global__ void gcn_deg_init(float* __restrict__ deg, int n) {
    int i = blockIdx.x * blockDim.x + threadIdx.x;
    if (i < n) deg[i] = FILL_WEIGHT;
}

// 2) deg[row[e]] += w[e]
__global__ void gcn_deg_accum(const int* __restrict__ row,
                              const float* __restrict__ w,
                              float* __restrict__ deg, int ne) {
    int e = blockIdx.x * blockDim.x + threadIdx.x;
    if (e < ne) atomicAdd(deg + row[e], w[e]);
}

// 3) dinv[i] = deg>0 ? rsqrt(max(deg,1e-12)) : 0
__global__ void gcn_dinv(const float* __restrict__ deg,
                         float* __restrict__ dinv, int n) {
    int i = blockIdx.x * blockDim.x + threadIdx.x;
    if (i < n) {
        float d = deg[i];
        dinv[i] = (d > 0.0f) ? __frsqrt_rn(fmaxf(d, 1e-12f)) : 0.0f;
    }
}

// ---------------------------------------------------------------------------
// 4) h = x @ kernel  via V_WMMA_F32_16X16X4_F32 (fp32-exact path).
// One wave -> one 16-row tile of h (16x64 output = 4 col-tiles of 16x16).
// f32 WMMA VGPR layouts (ISA 7.12.2):
//   A 16x4 : lane(m=lane&15, half=lane>>4): v0 = A[m][2*half], v1 = A[m][2*half+1]
//   B 4x16 : v0 = B[2*half][n],  v1 = B[2*half+1][n]   (n = lane&15)
//   C/D 16x16: VGPR v holds M = v + 8*half, N = lane&15
__global__ __launch_bounds__(256)
void gcn_gemm_wmma(const float* __restrict__ x,     // [N,64]
                   const float* __restrict__ kern,  // [64,64]
                   float* __restrict__ h,           // [N,64]
                   int num_nodes) {
    const int lane  = threadIdx.x & 31;
    const int wave  = threadIdx.x >> 5;
    const int tile  = blockIdx.x * 8 + wave;          // 16-row tile index
    const int ntile = (num_nodes + 15) >> 4;
    if (tile >= ntile) return;                        // wave-uniform guard

    const int r0   = tile << 4;
    const int half = lane >> 4;                       // 0 or 1
    const int m    = lane & 15;

    v8f acc[4] = {};                                  // 4 col-tiles of 16x16 f32

    const float* xrow = x + (size_t)(r0 + m) * FEAT;
#pragma unroll
    for (int kb = 0; kb < FEAT; kb += 4) {
        const int k0 = kb + (half << 1);
        v2f a;
        a.x = xrow[k0];
        a.y = xrow[k0 + 1];
#pragma unroll
        for (int cb = 0; cb < 4; ++cb) {
            v2f b;
            b.x = kern[(size_t)k0 * FEAT + cb * 16 + m];
            b.y = kern[(size_t)(k0 + 1) * FEAT + cb * 16 + m];
            acc[cb] = __builtin_amdgcn_wmma_f32_16x16x4_f32(
                /*neg_a=*/false, a, /*neg_b=*/false, b,
                /*c_mod=*/(short)0, acc[cb],
                /*reuse_a=*/false, /*reuse_b=*/false);
        }
    }

#pragma unroll
    for (int cb = 0; cb < 4; ++cb) {
#pragma unroll
        for (int v = 0; v < 8; ++v) {
            h[(size_t)(r0 + v + 8 * half) * FEAT + cb * 16 + m] = acc[cb][v];
        }
    }
}

// ---------------------------------------------------------------------------
// 5) out[i,f] = bias[f] + h[i,f] * dinv[i]^2 * fill_weight   (self-loop + bias)
__global__ void gcn_out_init(const float* __restrict__ h,
                             const float* __restrict__ dinv,
                             const float* __restrict__ bias,
                             float* __restrict__ out, long long total) {
    long long idx = (long long)blockIdx.x * blockDim.x + threadIdx.x;
    if (idx >= total) return;
    int i = (int)(idx >> 6);
    int f = (int)(idx & 63);
    float d = dinv[i];
    out[idx] = bias[f] + h[idx] * d * d * FILL_WEIGHT;
}

// 6) out[row,f] += h[col,f] * dinv[row]*w*dinv[col]
//    64 consecutive threads per edge -> row/col/w are wave-uniform (scalarize);
//    per-lane fp32 global atomics land in L2 (h is 25.6 MB, L2-resident).
__global__ void gcn_edge_scatter(const int* __restrict__ row,
                                 const int* __restrict__ col,
                                 const float* __restrict__ w,
                                 const float* __restrict__ dinv,
                                 const float* __restrict__ h,
                                 float* __restrict__ out, long long total) {
    long long idx = (long long)blockIdx.x * blockDim.x + threadIdx.x;
    if (idx >= total) return;
    int e = (int)(idx >> 6);
    int f = (int)(idx & 63);
    int r = row[e];
    int c = col[e];
    float wn = dinv[r] * w[e] * dinv[c];
    atomicAdd(out + (size_t)r * FEAT + f, h[(size_t)c * FEAT + f] * wn);
}

// ---------------------------------------------------------------------------
extern "C" void kernel_launch(void* const* d_in, const int* in_sizes, int n_in,
                              void* d_out, int out_size, void* d_ws, size_t ws_size,
                              hipStream_t stream) {
    const float* x    = (const float*)d_in[0];   // [N,64]
    const int*   ei   = (const int*)d_in[1];     // [2,E]
    const float* ew   = (const float*)d_in[2];   // [E]
    const float* kern = (const float*)d_in[3];   // [64,64]
    const float* bias = (const float*)d_in[4];   // [64]

    const int N = in_sizes[0] / FEAT;
    const int E = in_sizes[2];
    const int* row = ei;        // edge_index[0]
    const int* col = ei + E;    // edge_index[1]

    // workspace layout: h [N*64] | deg [N] | dinv [N]
    float* h    = (float*)d_ws;
    float* deg  = h + (size_t)N * FEAT;
    float* dinv = deg + N;
    float* out  = (float*)d_out;

    const int T = 256;

    gcn_deg_init<<<(N + T - 1) / T, T, 0, stream>>>(deg, N);
    gcn_deg_accum<<<(E + T - 1) / T, T, 0, stream>>>(row, ew, deg, E);
    gcn_dinv<<<(N + T - 1) / T, T, 0, stream>>>(deg, dinv, N);

    const int ntile = (N + 15) / 16;                    // 6250 wave-tiles
    gcn_gemm_wmma<<<(ntile + 7) / 8, T, 0, stream>>>(x, kern, h, N);

    long long tn = (long long)N * FEAT;
    gcn_out_init<<<(int)((tn + T - 1) / T), T, 0, stream>>>(h, dinv, bias, out, tn);

    long long te = (long long)E * FEAT;
    gcn_edge_scatter<<<(int)((te + T - 1) / T), T, 0, stream>>>(row, col, ew, dinv, h, out, te);
}